// BiLSTM_CRF_34754875359932
// MI455X (gfx1250) — compile-verified
//
#include <hip/hip_runtime.h>
#include <hip/hip_bf16.h>

// CRF NLL on MI455X (gfx1250).
// Forward recurrence rewritten in real space so the per-step 64x256 batched
// mat-vec maps onto V_WMMA_F32_16X16X4_F32 (fp32 WMMA -> no precision drift
// over the 512-step scan).

typedef __attribute__((ext_vector_type(2))) float v2f;
typedef __attribute__((ext_vector_type(8))) float v8f;

#define BSZ   256
#define SSZ   512
#define TSZ   50      // real tag count
#define TP    64      // padded tag count (4 x 16 WMMA tiles)
#define COLS  16      // sentences per workgroup (one WMMA N-tile)
#define START 48
#define STOP  49
#define NEGV  (-10000.0f)

// ---------------------------------------------------------------------------
// Gold path score: trivially sequential per sentence, one thread per sentence.
// ---------------------------------------------------------------------------
__global__ __launch_bounds__(128) void crf_gold_kernel(
    const float* __restrict__ feats, const int* __restrict__ tags,
    const float* __restrict__ trans, float* __restrict__ gold)
{
  int b = blockIdx.x * blockDim.x + threadIdx.x;
  if (b >= BSZ) return;
  int prev = START;
  float acc = 0.0f;
  const int* trow = tags + (size_t)b * SSZ;
  const float* fbase = feats + (size_t)b * SSZ * TSZ;
  for (int s = 0; s < SSZ; ++s) {
    int tag = trow[s];
    if (tag != 1) {                       // tag==1 => masked/deleted timestep
      acc += trans[tag * TSZ + prev] + fbase[(size_t)s * TSZ + tag];
      prev = tag;
    }
  }
  gold[b] = acc + trans[STOP * TSZ + prev];
}

// ---------------------------------------------------------------------------
// Forward algorithm (log partition) via per-step WMMA GEMM:
//   new_fv = feat_s + m + log( exp(trans) @ exp(fv - m) )
// One workgroup = 16 sentences (columns). 4 waves, each owns one 16-row tile
// of the padded 64x16 state. E = exp(trans) is loop-invariant -> A fragments
// preloaded in VGPRs.
// ---------------------------------------------------------------------------
__global__ __launch_bounds__(128) void crf_forward_kernel(
    const float* __restrict__ feats, const int* __restrict__ tags,
    const float* __restrict__ trans, const float* __restrict__ gold,
    float* __restrict__ out)
{
  __shared__ float Epad[TP][TP];     // exp(transitions), zero-padded
  __shared__ float tstop[TP];        // transitions[STOP][*]
  __shared__ float fv[TP][COLS];     // forward variable (log space)
  __shared__ float fvexp[TP][COLS];  // exp(fv - m) per column
  __shared__ float mcol[COLS];       // per-column max
  __shared__ int   maskcol[COLS];    // per-column "tag==1" mask for step s

  const int tid  = threadIdx.x;
  const int lane = tid & 31;
  const int wv   = tid >> 5;                 // wave id 0..3 -> row tile
  const int colbase = blockIdx.x * COLS;

  // Build E = exp(trans) padded to 64x64 (pad 'from' columns with 0 so padded
  // rows of X never contribute; padded 'to' rows are ignored on writeback).
  for (int i = tid; i < TP * TP; i += 128) {
    int to = i >> 6, from = i & 63;
    float e = 0.0f;
    if (to < TSZ && from < TSZ) e = __expf(trans[to * TSZ + from]);
    Epad[to][from] = e;
  }
  for (int i = tid; i < TP; i += 128)
    tstop[i] = (i < TSZ) ? trans[STOP * TSZ + i] : NEGV;
  for (int i = tid; i < TP * COLS; i += 128) {
    int to = i >> 4, c = i & 15;
    fv[to][c] = (to == START) ? 0.0f : NEGV;
  }
  __syncthreads();

  // --- preload A fragments of E for this wave's 16-row tile ---------------
  // A-matrix 16x4 layout: lanes 0-15 -> M=lane, K={0,1}; lanes 16-31 -> K={2,3}
  const int M  = lane & 15;
  const int kb = (lane < 16) ? 0 : 2;
  const int n  = lane & 15;                  // output column (N) for C/D
  const int hi = (lane < 16) ? 0 : 8;        // C/D: high lanes hold M+8
  float a0[16], a1[16];
#pragma unroll
  for (int k = 0; k < 16; ++k) {
    a0[k] = Epad[wv * 16 + M][4 * k + kb];
    a1[k] = Epad[wv * 16 + M][4 * k + kb + 1];
  }

  // --- sequential scan over timesteps -------------------------------------
#pragma unroll 1
  for (int s = 0; s < SSZ; ++s) {
    // phase 1: per-column max + mask bit
    if (tid < COLS) {
      float m = NEGV;
      for (int to = 0; to < TP; ++to) m = fmaxf(m, fv[to][tid]);
      mcol[tid] = m;
      maskcol[tid] = (tags[(size_t)(colbase + tid) * SSZ + s] == 1) ? 1 : 0;
    }
    __syncthreads();

    // phase 2: X = exp(fv - m)  (padded rows: exp(-10000-m) == 0 exactly)
    for (int i = tid; i < TP * COLS; i += 128) {
      int to = i >> 4, c = i & 15;
      fvexp[to][c] = __expf(fv[to][c] - mcol[c]);
    }
    __syncthreads();

    // phase 3: D(16x16) = E_tile(16x64) @ X(64x16) via 16 chained f32 WMMAs
    v8f acc = {0.0f, 0.0f, 0.0f, 0.0f, 0.0f, 0.0f, 0.0f, 0.0f};
#pragma unroll
    for (int k = 0; k < 16; ++k) {
      v2f Af = { a0[k], a1[k] };
      v2f Bf = { fvexp[4 * k + kb][n], fvexp[4 * k + kb + 1][n] };
      acc = __builtin_amdgcn_wmma_f32_16x16x4_f32(
          /*neg_a=*/false, Af, /*neg_b=*/false, Bf,
          /*c_mod=*/(short)0, acc, /*reuse_a=*/false, /*reuse_b=*/false);
    }

    // phase 4: fv update for this wave's rows (skip padded rows / masked cols)
    {
      const int   b    = colbase + n;
      const float m    = mcol[n];
      const int   msk  = maskcol[n];
      const float* frow = feats + ((size_t)b * SSZ + s) * TSZ;
#pragma unroll
      for (int v = 0; v < 8; ++v) {
        int to = wv * 16 + hi + v;           // C/D layout: M = v (+8 high half)
        if (to < TSZ && !msk) {
          fv[to][n] = frow[to] + m + __logf(acc[v]);
        }
      }
    }
    __syncthreads();
  }

  // --- epilogue: fwd_score = logsumexp(fv + trans[STOP]); out = fwd - gold --
  if (tid < COLS) {
    float m = NEGV;
    for (int to = 0; to < TSZ; ++to) m = fmaxf(m, fv[to][tid] + tstop[to]);
    float ssum = 0.0f;
    for (int to = 0; to < TSZ; ++to) ssum += __expf(fv[to][tid] + tstop[to] - m);
    float fwd = m + __logf(ssum);
    out[colbase + tid] = fwd - gold[colbase + tid];
  }
}

// ---------------------------------------------------------------------------
extern "C" void kernel_launch(void* const* d_in, const int* in_sizes, int n_in,
                              void* d_out, int out_size, void* d_ws, size_t ws_size,
                              hipStream_t stream) {
  (void)in_sizes; (void)n_in; (void)out_size; (void)ws_size;
  const float* feats = (const float*)d_in[0];   // [B,S,T] f32
  const int*   tags  = (const int*)d_in[1];     // [B,S]   i32
  const float* trans = (const float*)d_in[2];   // [T,T]   f32
  float* out  = (float*)d_out;                  // [B]     f32
  float* gold = (float*)d_ws;                   // [B] scratch

  crf_gold_kernel<<<2, 128, 0, stream>>>(feats, tags, trans, gold);
  crf_forward_kernel<<<BSZ / COLS, 128, 0, stream>>>(feats, tags, trans, gold, out);
}